// GridRNN_69002944577694
// MI455X (gfx1250) — compile-verified
//
#include <hip/hip_runtime.h>
#include <hip/hip_bf16.h>
#include <math.h>

typedef float v2f __attribute__((ext_vector_type(2)));
typedef float v8f __attribute__((ext_vector_type(8)));

#define Hdim 256
#define GRIDN 128   // S == T == 128

// ---------------------------------------------------------------------------
// Parallel input-side GEMM:  out[r,n] = sum_k in[r,k]*W[n,k] + b0[n] + b1[n]
// One wave computes one 16x16 tile with V_WMMA_F32_16X16X4_F32, K-loop of 64.
// ---------------------------------------------------------------------------
__global__ __launch_bounds__(256) void u_gemm_kernel(
    const float* __restrict__ in,   // [R, 256]
    const float* __restrict__ w,    // [256, 256], row n holds W[n, k]
    const float* __restrict__ b0,   // [256]
    const float* __restrict__ b1,   // [256]
    float* __restrict__ out,        // [R, 256]
    int mtiles)                     // R / 16
{
    const int wave = blockIdx.x * (blockDim.x >> 5) + (threadIdx.x >> 5);
    const int lane = threadIdx.x & 31;
    const int mt = wave >> 4;         // 16 N-tiles per M-tile row
    const int nt = wave & 15;
    if (mt >= mtiles) return;         // uniform per-wave

    const int mh = lane & 15;         // A: M index / B: N index
    const int kb = (lane >> 4) << 1;  // K sub-offset {0,2}
    const int hi = lane >> 4;
    const int n  = nt * 16 + mh;

    const float* arow = in + (mt * 16 + mh) * Hdim + kb;
    const float* brow = w  + n * Hdim + kb;

    const float bias = b0[n] + b1[n];
    v8f c;
#pragma unroll
    for (int v = 0; v < 8; ++v) c[v] = bias;

#pragma unroll 8
    for (int k0 = 0; k0 < Hdim; k0 += 4) {
        v2f a = *(const v2f*)(arow + k0);
        v2f b = *(const v2f*)(brow + k0);
        c = __builtin_amdgcn_wmma_f32_16x16x4_f32(false, a, false, b,
                                                  (short)0, c, false, false);
    }

#pragma unroll
    for (int v = 0; v < 8; ++v)
        out[(mt * 16 + v + 8 * hi) * Hdim + n] = c[v];
}

// ---------------------------------------------------------------------------
// Recurrence kernel. 32 blocks x 512 threads (16 waves). Each block owns 16
// independent chains (mode 0: wrapping-diagonal chains of h_x; mode 1: row
// chains of h_y). State h[16][256] lives in LDS (row stride 260 dwords ->
// conflict-free fragment reads). Per step: D[16,256] = h @ Whh^T with fp32
// WMMA; Whh fragments are hoisted into registers (step-invariant, 128 VGPRs)
// so the sequential inner loop touches only LDS + XDL. U loads issue before
// the WMMA chain; next step's U line is prefetched.
// ---------------------------------------------------------------------------
__global__ __launch_bounds__(512) void rec_kernel(
    const float* __restrict__ U,    // element (b,i,j,h) at (b*us+i*ui+j*uj)*256+h
    const float* __restrict__ whh,  // [256,256], row n holds Whh[n,k]
    float* __restrict__ out,        // element (b,i,j,h) at ((b*128+i)*128+j)*ps+so+h
    int us, int ui, int uj,
    int ps, int so,
    int mode)                       // 0 = x (diagonal), 1 = y (rows)
{
    __shared__ float hbuf[16 * 260];

    const int tid  = threadIdx.x;
    const int wave = tid >> 5;
    const int lane = tid & 31;
    const int mh   = lane & 15;
    const int hi   = lane >> 4;
    const int kb   = hi << 1;
    const int n    = wave * 16 + mh;
    const int grp  = blockIdx.x;          // chain group (16 chains)

    for (int idx = tid; idx < 16 * 260; idx += 512) hbuf[idx] = 0.0f;

    // ---- hoist the step-invariant B matrix (Whh^T fragments) into VGPRs ----
    const float* brow = whh + n * Hdim + kb;
    v2f bfrag[64];
#pragma unroll
    for (int kk = 0; kk < 64; ++kk)
        bfrag[kk] = *(const v2f*)(brow + 4 * kk);

    // per-v chain decomposition (step-invariant)
    int bb[8], rr[8];
#pragma unroll
    for (int v = 0; v < 8; ++v) {
        const int chain = grp * 16 + v + 8 * hi;
        bb[v] = chain >> 7;
        rr[v] = chain & 127;
    }

    const float* afrag = hbuf + mh * 260 + kb;
    __syncthreads();

    for (int step = 0; step < GRIDN; ++step) {
        // ---- issue U loads early; prefetch next step's U line ----
        float uval[8];
        int   ooff[8];
#pragma unroll
        for (int v = 0; v < 8; ++v) {
            int i, j;
            if (mode == 0) { i = step; j = (step + rr[v]) & 127; }
            else           { i = rr[v]; j = step; }
            const int uoff = (bb[v] * us + i * ui + j * uj) * Hdim + n;
            uval[v] = U[uoff];
            ooff[v] = ((bb[v] * GRIDN + i) * GRIDN + j) * ps + so + n;
            if (step + 1 < GRIDN) {
                int i2, j2;
                if (mode == 0) { i2 = step + 1; j2 = (step + 1 + rr[v]) & 127; }
                else           { i2 = rr[v];    j2 = step + 1; }
                __builtin_prefetch(U + (bb[v] * us + i2 * ui + j2 * uj) * Hdim + n, 0, 0);
            }
        }

        // ---- D[16,256] += h_prev @ Whh^T : dual accumulator chains ----
        v8f c0 = {};
        v8f c1 = {};
#pragma unroll
        for (int kk = 0; kk < 64; kk += 2) {
            v2f a0 = *(const v2f*)(afrag + 4 * kk);
            c0 = __builtin_amdgcn_wmma_f32_16x16x4_f32(false, a0, false, bfrag[kk],
                                                       (short)0, c0, false, false);
            v2f a1 = *(const v2f*)(afrag + 4 * (kk + 1));
            c1 = __builtin_amdgcn_wmma_f32_16x16x4_f32(false, a1, false, bfrag[kk + 1],
                                                       (short)0, c1, false, false);
        }

        float hnew[8];
#pragma unroll
        for (int v = 0; v < 8; ++v) {
            const float val = tanhf(c0[v] + c1[v] + uval[v]);
            out[ooff[v]] = val;
            hnew[v] = val;
        }

        __syncthreads();   // all waves done reading previous state
#pragma unroll
        for (int v = 0; v < 8; ++v)
            hbuf[(v + 8 * hi) * 260 + n] = hnew[v];
        __syncthreads();   // new state visible
    }
}

// ---------------------------------------------------------------------------
extern "C" void kernel_launch(void* const* d_in, const int* in_sizes, int n_in,
                              void* d_out, int out_size, void* d_ws, size_t ws_size,
                              hipStream_t stream) {
    (void)in_sizes; (void)n_in; (void)out_size; (void)ws_size;

    const float* src   = (const float*)d_in[0];   // [4,128,256]
    const float* trg   = (const float*)d_in[1];   // [4,128,256]
    const float* Wx_ih = (const float*)d_in[2];   // [3,256,256]
    const float* Wx_hh = (const float*)d_in[3];
    const float* bx_ih = (const float*)d_in[4];   // [3,256]
    const float* bx_hh = (const float*)d_in[5];
    const float* Wy_ih = (const float*)d_in[6];
    const float* Wy_hh = (const float*)d_in[7];
    const float* by_ih = (const float*)d_in[8];
    const float* by_hh = (const float*)d_in[9];

    float* out = (float*)d_out;
    float* ws  = (float*)d_ws;

    const long GRID = 4L * 128 * 128 * 256;       // 16,777,216 floats
    float* hx = ws;
    float* hy = ws + GRID;
    float* Ux = ws + 2 * GRID;
    float* Uy = ws + 3 * GRID;

    const int HH = Hdim * Hdim;

    for (int d = 0; d < 3; ++d) {
        // ----- input-side GEMMs (fully parallel) -----
        const float *xin, *yin;
        int R;
        if (d == 0) { xin = src; yin = trg; R = 4 * 128; }       // broadcast inputs
        else        { xin = hx;  yin = hy;  R = 4 * 128 * 128; }
        const int mtiles = R / 16;
        const int blocks = (mtiles * 16 + 7) / 8;   // 8 waves (tiles) per block

        u_gemm_kernel<<<blocks, 256, 0, stream>>>(
            xin, Wx_ih + d * HH, bx_ih + d * Hdim, bx_hh + d * Hdim, Ux, mtiles);
        u_gemm_kernel<<<blocks, 256, 0, stream>>>(
            yin, Wy_ih + d * HH, by_ih + d * Hdim, by_hh + d * Hdim, Uy, mtiles);

        // ----- U addressing (full grid vs broadcast at depth 0) -----
        int usx, uix, ujx, usy, uiy, ujy;
        if (d == 0) {
            usx = 128; uix = 1; ujx = 0;     // Ux[b,i]
            usy = 128; uiy = 0; ujy = 1;     // Uy[b,j]
        } else {
            usx = usy = 128 * 128; uix = uiy = 128; ujx = ujy = 1;  // full grid
        }

        // ----- output targets: ws grids for d<2, interleaved d_out for d==2 -----
        float* ox; float* oy; int ps, sox, soy;
        if (d == 2) { ox = out; oy = out; ps = 2 * Hdim; sox = 0; soy = Hdim; }
        else        { ox = hx;  oy = hy;  ps = Hdim;     sox = 0; soy = 0; }

        // ----- sequential recurrences: 512 chains, 16 per block -----
        rec_kernel<<<32, 512, 0, stream>>>(Ux, Wx_hh + d * HH, ox,
                                           usx, uix, ujx, ps, sox, /*mode=*/0);
        rec_kernel<<<32, 512, 0, stream>>>(Uy, Wy_hh + d * HH, oy,
                                           usy, uiy, ujy, ps, soy, /*mode=*/1);
    }
}